// MambaBlock_77781857731234
// MI455X (gfx1250) — compile-verified
//
#include <hip/hip_runtime.h>
#include <hip/hip_bf16.h>

// ---------------- Mamba block constants (match reference) ----------------
#define BATCH    4
#define LSEQ     1024
#define DMODEL   768
#define DINNER   1536          // EXPAND * DMODEL
#define DTRANK   48            // (768+15)//16
#define DSTATE   16
#define NROWS    (BATCH * LSEQ)        // 4096
#define XPROJ_N  (DTRANK + 2 * DSTATE) // 80
#define LN_EPS   1e-5f

typedef float v2f __attribute__((ext_vector_type(2)));
typedef float v8f __attribute__((ext_vector_type(8)));

// ---------------- 1) LayerNorm over channels (per (b,l) row) ----------------
// x is (B, DMODEL, L); hn written row-major (B*L, DMODEL)
__global__ __launch_bounds__(256) void ln_kernel(
    const float* __restrict__ x, const float* __restrict__ gamma,
    const float* __restrict__ beta, float* __restrict__ hn)
{
    const int row = blockIdx.x;          // b*L + l
    const int b = row >> 10;
    const int l = row & (LSEQ - 1);
    const int tid = threadIdx.x;
    const float* xb = x + (size_t)b * DMODEL * LSEQ + l;  // stride LSEQ per channel

    float v0 = xb[(size_t)(tid      ) * LSEQ];
    float v1 = xb[(size_t)(tid + 256) * LSEQ];
    float v2 = xb[(size_t)(tid + 512) * LSEQ];

    __shared__ float red[256];
    red[tid] = v0 + v1 + v2;
    __syncthreads();
    for (int off = 128; off > 0; off >>= 1) {
        if (tid < off) red[tid] += red[tid + off];
        __syncthreads();
    }
    const float mu = red[0] * (1.0f / DMODEL);
    __syncthreads();

    const float d0 = v0 - mu, d1 = v1 - mu, d2 = v2 - mu;
    red[tid] = d0 * d0 + d1 * d1 + d2 * d2;
    __syncthreads();
    for (int off = 128; off > 0; off >>= 1) {
        if (tid < off) red[tid] += red[tid + off];
        __syncthreads();
    }
    const float var = red[0] * (1.0f / DMODEL);
    const float rs = rsqrtf(var + LN_EPS);

    float* out = hn + (size_t)row * DMODEL;
    out[tid      ] = d0 * rs * gamma[tid      ] + beta[tid      ];
    out[tid + 256] = d1 * rs * gamma[tid + 256] + beta[tid + 256];
    out[tid + 512] = d2 * rs * gamma[tid + 512] + beta[tid + 512];
}

// ---------------- 2) FP32 WMMA GEMM: Out[M,N] = Act[M,K(lda)] * W[N,K]^T ----------------
// Double-buffered async global->LDS staging (GLOBAL_LOAD_ASYNC_TO_LDS_B128 +
// s_wait_asynccnt), fp32 WMMA 16x16x4 compute.
// mode 0: plain store to Out (ld = ldo)
// mode 1: Out = softplus(acc + bias[n])                        (dt path)
// mode 2: d_out[b, n, l] = acc + resid[b, n, l]   (m = b*L+l)  (final proj)
#define ROWSTRIDE 20                       // floats; 80B rows -> b128-aligned, conflict-free
#define TILE_F    (64 * ROWSTRIDE)         // 1280 floats per 64x16 slab
#define BUF_F     (2 * TILE_F)             // A tile + B tile per buffer

__global__ __launch_bounds__(256) void wmma_gemm_kernel(
    const float* __restrict__ Act, int lda,
    const float* __restrict__ Wt,            // row-major [N][K]
    float* __restrict__ Out, int ldo,
    int M, int N, int K,
    const float* __restrict__ bias,
    const float* __restrict__ resid,
    int mode)
{
    __shared__ float smem[2 * BUF_F];        // two staging buffers

    const int tid  = threadIdx.x;
    const int wave = tid >> 5;
    const int lane = tid & 31;
    const int m0 = blockIdx.y * 64;
    const int n0 = blockIdx.x * 64;
    const int sm  = (wave >> 1) * 16;        // wave's sub-tile row
    const int sn0 = (wave & 1) * 32;         // wave's first sub-tile col (owns two)

    v8f acc0 = {};
    v8f acc1 = {};

    const int lr = tid >> 2;                 // 0..63 : row within 64-row tile
    const int lc = (tid & 3) * 4;            // 0,4,8,12 : col within 16-wide K slab
    const int fr = lane & 15;                // fragment row (M or N within sub-tile)
    const int kh = (lane >> 4) * 2;          // lanes 16-31 hold K+2,K+3 per ISA layout

    // ragged-N (N=80) handled by clamping: garbage cols are never stored
    const int wrow = (n0 + lr < N) ? (n0 + lr) : (N - 1);
    const float* aBase = Act + (size_t)(m0 + lr) * lda + lc;
    const float* bBase = Wt  + (size_t)wrow * K + lc;

    // LDS byte addresses for the async destinations, derived FROM the smem
    // pointer itself: low 32 bits of the flat address == wave-relative LDS
    // offset (shared aperture is entirely in addr[63:32]), and the ptrtoint
    // capture makes smem escape so the "memory"-clobbered asm is known to
    // write it (keeps the ds_load fragment reads alive).
    const int stage = lr * ROWSTRIDE + lc;
    const unsigned aAddr0 = (unsigned)(size_t)&smem[stage];
    const unsigned bAddr0 = (unsigned)(size_t)&smem[TILE_F + stage];
    const unsigned aAddr1 = (unsigned)(size_t)&smem[BUF_F + stage];
    const unsigned bAddr1 = (unsigned)(size_t)&smem[BUF_F + TILE_F + stage];

    const int nk = K >> 4;                   // K is always a multiple of 16 here

    // prologue: stage slab 0 into buffer 0 (each lane copies 16B straight to LDS)
    asm volatile("global_load_async_to_lds_b128 %0, %1, off"
                 :: "v"(aAddr0), "v"(aBase) : "memory");
    asm volatile("global_load_async_to_lds_b128 %0, %1, off"
                 :: "v"(bAddr0), "v"(bBase) : "memory");

    for (int k = 0; k < nk; ++k) {
        if (k + 1 < nk) {
            // stage slab k+1 into the other buffer, then wait for slab k
            const unsigned aA = ((k + 1) & 1) ? aAddr1 : aAddr0;
            const unsigned bA = ((k + 1) & 1) ? bAddr1 : bAddr0;
            asm volatile("global_load_async_to_lds_b128 %0, %1, off"
                         :: "v"(aA), "v"(aBase + (size_t)(k + 1) * 16) : "memory");
            asm volatile("global_load_async_to_lds_b128 %0, %1, off"
                         :: "v"(bA), "v"(bBase + (size_t)(k + 1) * 16) : "memory");
            asm volatile("s_wait_asynccnt 0x2" ::: "memory");  // slab k retired (in-order)
        } else {
            asm volatile("s_wait_asynccnt 0x0" ::: "memory");
        }
        __syncthreads();

        const float* As = smem + (size_t)(k & 1) * BUF_F;      // 64 x 16 (stride 20)
        const float* Bs = As + TILE_F;
#pragma unroll
        for (int kk = 0; kk < 4; ++kk) {
            v2f a, b0, b1;
            a[0]  = As[(sm + fr) * ROWSTRIDE + kk * 4 + kh];
            a[1]  = As[(sm + fr) * ROWSTRIDE + kk * 4 + kh + 1];
            b0[0] = Bs[(sn0 + fr) * ROWSTRIDE + kk * 4 + kh];
            b0[1] = Bs[(sn0 + fr) * ROWSTRIDE + kk * 4 + kh + 1];
            b1[0] = Bs[(sn0 + 16 + fr) * ROWSTRIDE + kk * 4 + kh];
            b1[1] = Bs[(sn0 + 16 + fr) * ROWSTRIDE + kk * 4 + kh + 1];
            acc0 = __builtin_amdgcn_wmma_f32_16x16x4_f32(
                false, a, false, b0, (short)0, acc0, false, false);
            acc1 = __builtin_amdgcn_wmma_f32_16x16x4_f32(
                false, a, false, b1, (short)0, acc1, false, false);
        }
        __syncthreads();   // all waves done reading buffer (k&1) before it is re-staged
    }

    // epilogue: C/D layout -> VGPR v: lanes0-15 M=v, lanes16-31 M=8+v; N=lane&15
    const int rh = (lane >> 4) * 8;
#pragma unroll
    for (int v = 0; v < 8; ++v) {
        const int m = m0 + sm + rh + v;
        float vals[2] = { acc0[v], acc1[v] };
#pragma unroll
        for (int t = 0; t < 2; ++t) {
            const int n = n0 + sn0 + t * 16 + fr;
            if (m < M && n < N) {
                const float val = vals[t];
                if (mode == 0) {
                    Out[(size_t)m * ldo + n] = val;
                } else if (mode == 1) {
                    const float s = val + bias[n];
                    // stable softplus
                    Out[(size_t)m * ldo + n] =
                        (s > 20.f) ? s : __logf(1.f + __expf(s));
                } else {
                    const int bb = m >> 10, ll = m & (LSEQ - 1);
                    const size_t idx = ((size_t)bb * DMODEL + n) * LSEQ + ll;
                    Out[idx] = val + resid[idx];
                }
            }
        }
    }
}

// ---------------- 3) depthwise causal conv (width 4) + SiLU ----------------
__global__ __launch_bounds__(256) void conv_silu_kernel(
    const float* __restrict__ xz, const float* __restrict__ conv_w,
    const float* __restrict__ conv_b, float* __restrict__ x_ssm)
{
    const int idx = blockIdx.x * 256 + threadIdx.x;   // over B*L*DINNER
    const int d = idx % DINNER;
    const int l = (idx / DINNER) % LSEQ;
    const int b = idx / (DINNER * LSEQ);

    float acc = conv_b[d];
#pragma unroll
    for (int j = 0; j < 4; ++j) {
        const int ls = l - 3 + j;
        if (ls >= 0)
            acc += xz[((size_t)b * LSEQ + ls) * (2 * DINNER) + d] * conv_w[d * 4 + j];
    }
    const float s = acc / (1.f + __expf(-acc));       // SiLU
    x_ssm[((size_t)b * LSEQ + l) * DINNER + d] = s;
}

// ---------------- 4) selective scan, fused D-skip + SiLU(z) gating ----------------
// 16 lanes per (b,d) channel: lane = state index n. 4 shfl_xor hops reduce y.
__global__ __launch_bounds__(256) void scan_kernel(
    const float* __restrict__ dtb, const float* __restrict__ x_ssm,
    const float* __restrict__ x_dbl, const float* __restrict__ A_log,
    const float* __restrict__ Dp, const float* __restrict__ xz,
    float* __restrict__ yg)
{
    const int grp = blockIdx.x * 16 + (threadIdx.x >> 4);  // b*DINNER + d
    const int n = threadIdx.x & 15;
    const int b = grp / DINNER;
    const int d = grp % DINNER;

    const float An = -__expf(A_log[d * DSTATE + n]);
    const float Dd = Dp[d];
    float h = 0.f;

    for (int l = 0; l < LSEQ; ++l) {
        const size_t row = (size_t)b * LSEQ + l;
        const float dtv = dtb[row * DINNER + d];
        const float xv  = x_ssm[row * DINNER + d];
        const float Bv  = x_dbl[row * XPROJ_N + DTRANK + n];
        const float Cv  = x_dbl[row * XPROJ_N + DTRANK + DSTATE + n];

        const float dA = __expf(dtv * An);
        h = dA * h + (dtv * xv) * Bv;

        float p = h * Cv;
        p += __shfl_xor(p, 8, 16);
        p += __shfl_xor(p, 4, 16);
        p += __shfl_xor(p, 2, 16);
        p += __shfl_xor(p, 1, 16);

        if (n == 0) {
            float y = p + Dd * xv;
            const float z = xz[row * (2 * DINNER) + DINNER + d];
            y *= z / (1.f + __expf(-z));              // * SiLU(z)
            yg[row * DINNER + d] = y;
        }
    }
}

// ---------------- launch ----------------
extern "C" void kernel_launch(void* const* d_in, const int* in_sizes, int n_in,
                              void* d_out, int out_size, void* d_ws, size_t ws_size,
                              hipStream_t stream) {
    const float* x      = (const float*)d_in[0];
    const float* gamma  = (const float*)d_in[1];
    const float* beta   = (const float*)d_in[2];
    const float* W_in   = (const float*)d_in[3];   // (3072, 768)
    const float* conv_w = (const float*)d_in[4];   // (1536, 1, 4)
    const float* conv_b = (const float*)d_in[5];   // (1536)
    const float* W_xprj = (const float*)d_in[6];   // (80, 1536)
    const float* W_dt   = (const float*)d_in[7];   // (1536, 48)
    const float* b_dt   = (const float*)d_in[8];   // (1536)
    const float* A_log  = (const float*)d_in[9];   // (1536, 16)
    const float* Dp     = (const float*)d_in[10];  // (1536)
    const float* W_out  = (const float*)d_in[11];  // (768, 1536)

    float* ws    = (float*)d_ws;
    float* hn    = ws;                                   // 4096*768
    float* xz    = hn    + (size_t)NROWS * DMODEL;       // 4096*3072
    float* x_ssm = xz    + (size_t)NROWS * 2 * DINNER;   // 4096*1536
    float* x_dbl = x_ssm + (size_t)NROWS * DINNER;       // 4096*80
    float* dtb   = x_dbl + (size_t)NROWS * XPROJ_N;      // 4096*1536
    float* yg    = dtb   + (size_t)NROWS * DINNER;       // 4096*1536

    // 1) LayerNorm
    ln_kernel<<<NROWS, 256, 0, stream>>>(x, gamma, beta, hn);

    // 2) xz = hn @ W_in^T  : (4096,768) x (768,3072)
    {
        dim3 grid(2 * DINNER / 64, NROWS / 64);
        wmma_gemm_kernel<<<grid, 256, 0, stream>>>(
            hn, DMODEL, W_in, xz, 2 * DINNER,
            NROWS, 2 * DINNER, DMODEL, nullptr, nullptr, 0);
    }

    // 3) depthwise conv + SiLU
    conv_silu_kernel<<<(BATCH * LSEQ * DINNER) / 256, 256, 0, stream>>>(
        xz, conv_w, conv_b, x_ssm);

    // 4) x_dbl = x_ssm @ W_xproj^T : (4096,1536) x (1536,80)
    {
        dim3 grid((XPROJ_N + 63) / 64, NROWS / 64);
        wmma_gemm_kernel<<<grid, 256, 0, stream>>>(
            x_ssm, DINNER, W_xprj, x_dbl, XPROJ_N,
            NROWS, XPROJ_N, DINNER, nullptr, nullptr, 0);
    }

    // 5) dt = softplus(x_dbl[:, :48] @ W_dt^T + b_dt) : (4096,48) x (48,1536)
    {
        dim3 grid(DINNER / 64, NROWS / 64);
        wmma_gemm_kernel<<<grid, 256, 0, stream>>>(
            x_dbl, XPROJ_N, W_dt, dtb, DINNER,
            NROWS, DINNER, DTRANK, b_dt, nullptr, 1);
    }

    // 6) selective scan + D-skip + SiLU(z) gating
    scan_kernel<<<(BATCH * DINNER) / 16, 256, 0, stream>>>(
        dtb, x_ssm, x_dbl, A_log, Dp, xz, yg);

    // 7) out = yg @ W_out^T, transposed store + residual x
    {
        dim3 grid(DMODEL / 64, NROWS / 64);
        wmma_gemm_kernel<<<grid, 256, 0, stream>>>(
            yg, DINNER, W_out, (float*)d_out, 0,
            NROWS, DMODEL, DINNER, nullptr, x, 2);
    }
}